// AEGIS_89429809037959
// MI455X (gfx1250) — compile-verified
//
#include <hip/hip_runtime.h>
#include <math.h>

// Sizes from the reference: N=50000 nodes, E=600000 edges, D=128, H=8, DH=16.
#define DFEAT 128
#define NHEAD 8
#define DHEAD 16

typedef __attribute__((ext_vector_type(2))) float v2f;
typedef __attribute__((ext_vector_type(8))) float v8f;

// ---------------------------------------------------------------------------
// Order-preserving float <-> uint key so segment-max can use integer atomicMax
// ---------------------------------------------------------------------------
__device__ __forceinline__ unsigned int f32_key(float f) {
  unsigned int u = __float_as_uint(f);
  return (u & 0x80000000u) ? ~u : (u | 0x80000000u);
}
__device__ __forceinline__ float key_f32(unsigned int k) {
  unsigned int u = (k & 0x80000000u) ? (k & 0x7FFFFFFFu) : ~k;
  return __uint_as_float(u);
}

// ---------------------------------------------------------------------------
// Shared WMMA core: acc[4] += sA(64x128 tile) @ W(128x128), one wave computes
// 16 rows x 64 cols (4 WMMA col tiles). W staged in 4 K-chunks of 32 rows.
// Must be called by all 256 threads (contains __syncthreads).
// A frag 16x4 f32: v0/v1 = A[m][k0 + 2*half + {0,1}]
// B frag 4x16 f32: v0/v1 = W[k0 + 2*half + {0,1}][n]
// ---------------------------------------------------------------------------
__device__ __forceinline__ void mm64x128(
    const float* __restrict__ W, const float* sA, float* sW,
    int tid, int mrow, int ch, int half, int l15, v8f acc[4])
{
  for (int kc = 0; kc < 4; ++kc) {
    __syncthreads();                       // covers sA load / previous sW use
    for (int i = tid; i < 32 * DFEAT / 4; i += 256)
      ((float4*)sW)[i] = ((const float4*)W)[kc * (32 * DFEAT / 4) + i];
    __syncthreads();

    #pragma unroll
    for (int kt = 0; kt < 8; ++kt) {
      const int k0 = kt * 4 + half * 2;    // K offset within chunk (this lane)
      const int kg = kc * 32 + k0;         // global K for the A fragment
      v2f a;
      a.x = sA[mrow * DFEAT + kg];
      a.y = sA[mrow * DFEAT + kg + 1];
      #pragma unroll
      for (int ct = 0; ct < 4; ++ct) {
        const int ncol = ch * 64 + ct * 16 + l15;
        v2f b;
        b.x = sW[k0 * DFEAT + ncol];
        b.y = sW[(k0 + 1) * DFEAT + ncol];
        acc[ct] = __builtin_amdgcn_wmma_f32_16x16x4_f32(
            false, a, false, b, (short)0, acc[ct], false, false);
      }
    }
  }
}

__device__ __forceinline__ void stage_a_tile(
    const float* __restrict__ A, float* sA, int tid, int m0, int M)
{
  for (int i = tid; i < 64 * DFEAT / 4; i += 256) {
    const int r  = i >> 5;                 // 32 float4 per row
    const int c4 = i & 31;
    float4 val;
    if (m0 + r < M) val = ((const float4*)A)[(size_t)(m0 + r) * (DFEAT / 4) + c4];
    else            val = make_float4(0.f, 0.f, 0.f, 0.f);
    ((float4*)sA)[i] = val;
  }
}

// ---------------------------------------------------------------------------
// Fused Q/K/V projection: x tile staged once, three weight matrices applied.
// ---------------------------------------------------------------------------
__global__ __launch_bounds__(256) void qkv_gemm_wmma(
    const float* __restrict__ x,
    const float* __restrict__ WQ, const float* __restrict__ WK,
    const float* __restrict__ WV,
    float* __restrict__ q, float* __restrict__ k, float* __restrict__ v, int M)
{
  __shared__ float sA[64 * DFEAT];         // 32 KB
  __shared__ float sW[32 * DFEAT];         // 16 KB

  const int tid  = threadIdx.x;
  const int wave = tid >> 5;
  const int lane = tid & 31;
  const int half = lane >> 4;
  const int l15  = lane & 15;
  const int m0   = blockIdx.x * 64;
  const int rt   = wave >> 1;
  const int ch   = wave & 1;
  const int mrow = rt * 16 + l15;

  stage_a_tile(x, sA, tid, m0, M);

  const float* Ws[3] = {WQ, WK, WV};
  float*       Ys[3] = {q, k, v};

  for (int w = 0; w < 3; ++w) {
    v8f acc[4] = {};
    mm64x128(Ws[w], sA, sW, tid, mrow, ch, half, l15, acc);
    #pragma unroll
    for (int ct = 0; ct < 4; ++ct) {
      const int ncol = ch * 64 + ct * 16 + l15;
      #pragma unroll
      for (int i = 0; i < 8; ++i) {
        const int r = m0 + rt * 16 + half * 8 + i;
        if (r < M) Ys[w][(size_t)r * DFEAT + ncol] = acc[ct][i];
      }
    }
  }
}

// ---------------------------------------------------------------------------
// Fused edge pipeline: e = edge_attr @ WE (WMMA), then in the epilogue
//   ve[edge]    = v[src] + e                      (stored for aggregation)
//   score       = <q[dst], k[src]+e> * DH^-0.5    (cross-lane reduction)
//   mxkey[dst]  = atomicMax(key(score))           (segment max)
// Saves a full re-read of e plus the q/k gathers of a separate score pass.
// ---------------------------------------------------------------------------
__global__ __launch_bounds__(256) void edge_gemm_fused(
    const float* __restrict__ ea, const float* __restrict__ WE,
    const float* __restrict__ q, const float* __restrict__ k,
    const float* __restrict__ v,
    const int* __restrict__ src, const int* __restrict__ dst,
    float* __restrict__ vebuf, float* __restrict__ score,
    unsigned int* __restrict__ mxkey, int E)
{
  __shared__ float sA[64 * DFEAT];
  __shared__ float sW[32 * DFEAT];

  const int tid  = threadIdx.x;
  const int wave = tid >> 5;
  const int lane = tid & 31;
  const int half = lane >> 4;
  const int l15  = lane & 15;
  const int m0   = blockIdx.x * 64;
  const int rt   = wave >> 1;
  const int ch   = wave & 1;
  const int mrow = rt * 16 + l15;

  stage_a_tile(ea, sA, tid, m0, E);

  v8f acc[4] = {};
  mm64x128(WE, sA, sW, tid, mrow, ch, half, l15, acc);

  // Epilogue: wave covers heads (ch*4 .. ch*4+3), 16 edges (rows).
  #pragma unroll
  for (int ct = 0; ct < 4; ++ct) {
    const int h = ch * 4 + ct;                   // head index
    const int c = h * DHEAD + l15;               // feature column
    #pragma unroll
    for (int i = 0; i < 8; ++i) {
      const int r = m0 + rt * 16 + half * 8 + i; // edge index
      if (r < E) {
        const int s_ = src[r];
        const int d_ = dst[r];
        const float ev = acc[ct][i];
        const float kv = k[(size_t)s_ * DFEAT + c];
        const float qv = q[(size_t)d_ * DFEAT + c];
        const float vv = v[(size_t)s_ * DFEAT + c];
        vebuf[(size_t)r * DFEAT + c] = vv + ev;  // ve = v[src] + e
        float prod = qv * (kv + ev);
        // reduce across the 16-lane column group (xor masks stay in-half)
        #pragma unroll
        for (int m = 1; m < 16; m <<= 1) prod += __shfl_xor(prod, m, 32);
        if (l15 == 0) {
          const float sc = prod * 0.25f;         // DH^-0.5 = 1/sqrt(16)
          score[(size_t)r * NHEAD + h] = sc;
          atomicMax(&mxkey[d_ * NHEAD + h], f32_key(sc));
        }
      }
    }
  }
}

// ---------------------------------------------------------------------------
// Generic Y = A @ W (+bias) (+resid): used for out = agg@Wo + bo + x
// ---------------------------------------------------------------------------
__global__ __launch_bounds__(256) void gemm128_wmma(
    const float* __restrict__ A, const float* __restrict__ W,
    const float* __restrict__ bias, const float* __restrict__ resid,
    float* __restrict__ Y, int M)
{
  __shared__ float sA[64 * DFEAT];
  __shared__ float sW[32 * DFEAT];

  const int tid  = threadIdx.x;
  const int wave = tid >> 5;
  const int lane = tid & 31;
  const int half = lane >> 4;
  const int l15  = lane & 15;
  const int m0   = blockIdx.x * 64;
  const int rt   = wave >> 1;
  const int ch   = wave & 1;
  const int mrow = rt * 16 + l15;

  stage_a_tile(A, sA, tid, m0, M);

  v8f acc[4] = {};
  mm64x128(W, sA, sW, tid, mrow, ch, half, l15, acc);

  #pragma unroll
  for (int ct = 0; ct < 4; ++ct) {
    const int ncol = ch * 64 + ct * 16 + l15;
    #pragma unroll
    for (int i = 0; i < 8; ++i) {
      const int r = m0 + rt * 16 + half * 8 + i;
      if (r < M) {
        float vv = acc[ct][i];
        if (bias)  vv += bias[ncol];
        if (resid) vv += resid[(size_t)r * DFEAT + ncol];
        Y[(size_t)r * DFEAT + ncol] = vv;
      }
    }
  }
}

// ---------------------------------------------------------------------------
// ex = exp(score - mx[dst]); score <- ex; denom[dst] += ex
// ---------------------------------------------------------------------------
__global__ void edge_exp_sum(
    const int* __restrict__ dst, const unsigned int* __restrict__ mxkey,
    float* __restrict__ score, float* __restrict__ denom, int E)
{
  const int t = blockIdx.x * blockDim.x + threadIdx.x;
  if (t >= E * NHEAD) return;
  const int edge = t >> 3, h = t & 7;
  const int d = dst[edge];
  const float mx = key_f32(mxkey[d * NHEAD + h]);
  const float ex = expf(score[t] - mx);
  score[t] = ex;
  atomicAdd(&denom[d * NHEAD + h], ex);
}

// ---------------------------------------------------------------------------
// alpha = ex/denom[dst]; agg[dst] += alpha * ve[edge]
// ---------------------------------------------------------------------------
__global__ void edge_aggregate(
    const int* __restrict__ dst, const float* __restrict__ ve,
    const float* __restrict__ score, const float* __restrict__ denom,
    float* __restrict__ agg, int E)
{
  const int t = blockIdx.x * blockDim.x + threadIdx.x;
  if (t >= E * NHEAD) return;
  const int edge = t >> 3, h = t & 7;
  const int d = dst[edge];
  const float alpha = score[t] / denom[d * NHEAD + h];
  const float4* vp = (const float4*)(ve + (size_t)edge * DFEAT + h * DHEAD);
  float* ap = agg + (size_t)d * DFEAT + h * DHEAD;
  #pragma unroll
  for (int i = 0; i < 4; ++i) {
    const float4 vv = vp[i];
    atomicAdd(ap + i * 4 + 0, alpha * vv.x);
    atomicAdd(ap + i * 4 + 1, alpha * vv.y);
    atomicAdd(ap + i * 4 + 2, alpha * vv.z);
    atomicAdd(ap + i * 4 + 3, alpha * vv.w);
  }
}

// ---------------------------------------------------------------------------
// GraphNorm column statistics (column j handled by thread j of each block)
// ---------------------------------------------------------------------------
__global__ __launch_bounds__(128) void col_sum(
    const float* __restrict__ hbuf, float* __restrict__ colsum, int N, int rpb)
{
  const int c  = threadIdx.x;
  const int r0 = blockIdx.x * rpb;
  const int r1 = min(r0 + rpb, N);
  float s = 0.f;
  for (int r = r0; r < r1; ++r) s += hbuf[(size_t)r * DFEAT + c];
  atomicAdd(&colsum[c], s);
}

__global__ __launch_bounds__(128) void mean_finalize(
    const float* __restrict__ colsum, float* __restrict__ mean, int N)
{
  mean[threadIdx.x] = colsum[threadIdx.x] / (float)N;
}

__global__ __launch_bounds__(128) void col_sumsq(
    const float* __restrict__ hbuf, const float* __restrict__ mean,
    const float* __restrict__ gms, float* __restrict__ sumsq, int N, int rpb)
{
  const int c = threadIdx.x;
  const float mu = gms[c] * mean[c];
  const int r0 = blockIdx.x * rpb;
  const int r1 = min(r0 + rpb, N);
  float s = 0.f;
  for (int r = r0; r < r1; ++r) {
    const float hc = hbuf[(size_t)r * DFEAT + c] - mu;
    s += hc * hc;
  }
  atomicAdd(&sumsq[c], s);
}

__global__ __launch_bounds__(128) void invstd_finalize(
    const float* __restrict__ sumsq, const float* __restrict__ gw,
    float* __restrict__ wscale, int N)
{
  const int c = threadIdx.x;
  const float var = sumsq[c] / (float)N;
  wscale[c] = gw[c] * rsqrtf(var + 1e-5f);
}

// out = gelu_exact(wscale*(h - gms*mean) + gbias)
__global__ void final_gelu(
    const float* __restrict__ hbuf, const float* __restrict__ mean,
    const float* __restrict__ gms, const float* __restrict__ wscale,
    const float* __restrict__ gbias, float* __restrict__ out, int total)
{
  const int t = blockIdx.x * blockDim.x + threadIdx.x;
  if (t >= total) return;
  const int c = t & (DFEAT - 1);
  const float hc = hbuf[t] - gms[c] * mean[c];
  const float y  = wscale[c] * hc + gbias[c];
  out[t] = 0.5f * y * (1.f + erff(y * 0.70710678118654752f));
}

// ---------------------------------------------------------------------------
extern "C" void kernel_launch(void* const* d_in, const int* in_sizes, int n_in,
                              void* d_out, int out_size, void* d_ws, size_t ws_size,
                              hipStream_t stream)
{
  const float* x   = (const float*)d_in[0];
  const float* ea  = (const float*)d_in[1];
  const int*   ei  = (const int*)d_in[2];
  const float* WQ  = (const float*)d_in[3];
  const float* WK  = (const float*)d_in[4];
  const float* WV  = (const float*)d_in[5];
  const float* WE  = (const float*)d_in[6];
  const float* Wo  = (const float*)d_in[7];
  const float* bo  = (const float*)d_in[8];
  const float* gw  = (const float*)d_in[9];
  const float* gb  = (const float*)d_in[10];
  const float* gms = (const float*)d_in[11];

  const int N = in_sizes[0] / DFEAT;
  const int E = in_sizes[1] / DFEAT;
  const int* src = ei;
  const int* dst = ei + E;

  // carve workspace
  char* p = (char*)d_ws;
  auto carve = [&](size_t bytes) -> char* {
    char* r = p;
    p += (bytes + 255) & ~(size_t)255;
    return r;
  };
  float* q      = (float*)carve((size_t)N * DFEAT * 4);
  float* kbuf   = (float*)carve((size_t)N * DFEAT * 4);
  float* vbuf   = (float*)carve((size_t)N * DFEAT * 4);
  float* agg    = (float*)carve((size_t)N * DFEAT * 4);
  float* hbuf   = (float*)carve((size_t)N * DFEAT * 4);
  unsigned int* mxkey = (unsigned int*)carve((size_t)N * NHEAD * 4);
  float* denom  = (float*)carve((size_t)N * NHEAD * 4);
  float* colsum = (float*)carve(DFEAT * 4);
  float* sumsq  = (float*)carve(DFEAT * 4);
  float* mean   = (float*)carve(DFEAT * 4);
  float* wscale = (float*)carve(DFEAT * 4);
  float* score  = (float*)carve((size_t)E * NHEAD * 4);
  float* vebuf  = (float*)carve((size_t)E * DFEAT * 4);

  hipMemsetAsync(mxkey, 0, (size_t)N * NHEAD * 4, stream);
  hipMemsetAsync(denom, 0, (size_t)N * NHEAD * 4, stream);
  hipMemsetAsync(agg,   0, (size_t)N * DFEAT * 4, stream);
  hipMemsetAsync(colsum, 0, DFEAT * 4, stream);
  hipMemsetAsync(sumsq,  0, DFEAT * 4, stream);

  const int gN = (N + 63) / 64;
  const int gE = (E + 63) / 64;

  // fused Q/K/V projections (fp32-exact WMMA, x tile staged once)
  qkv_gemm_wmma<<<gN, 256, 0, stream>>>(x, WQ, WK, WV, q, kbuf, vbuf, N);

  // fused edge GEMM + ve + score + segment-max
  edge_gemm_fused<<<gE, 256, 0, stream>>>(ea, WE, q, kbuf, vbuf, src, dst,
                                          vebuf, score, mxkey, E);

  // softmax denominator, then weighted aggregation
  const int EH  = E * NHEAD;
  const int gEH = (EH + 255) / 256;
  edge_exp_sum  <<<gEH, 256, 0, stream>>>(dst, mxkey, score, denom, E);
  edge_aggregate<<<gEH, 256, 0, stream>>>(dst, vebuf, score, denom, agg, E);

  // out = agg@Wo + bo, fused residual (+x) -> h
  gemm128_wmma<<<gN, 256, 0, stream>>>(agg, Wo, bo, x, hbuf, N);

  // GraphNorm + exact GELU
  const int RPB = 256;
  const int gR  = (N + RPB - 1) / RPB;
  col_sum        <<<gR, 128, 0, stream>>>(hbuf, colsum, N, RPB);
  mean_finalize  <<<1,  128, 0, stream>>>(colsum, mean, N);
  col_sumsq      <<<gR, 128, 0, stream>>>(hbuf, mean, gms, sumsq, N, RPB);
  invstd_finalize<<<1,  128, 0, stream>>>(sumsq, gw, wscale, N);

  const int total = N * DFEAT;
  final_gelu<<<(total + 255) / 256, 256, 0, stream>>>(hbuf, mean, gms, wscale, gb,
                                                      (float*)d_out, total);
}